// TemporalFrequencyModulator_70111046140305
// MI455X (gfx1250) — compile-verified
//
#include <hip/hip_runtime.h>
#include <hip/hip_bf16.h>
#include <math.h>

typedef float v2f __attribute__((ext_vector_type(2)));
typedef float v8f __attribute__((ext_vector_type(8)));

#define TDIM 16
#define TEXT_UNROLL 768

// ---------------------------------------------------------------------------
// Kernel 1: per-(b,c) fused operator  M = D^T * diag(sigmoid(E@Wf^T+bf)) * D
// One block per (b,c); 256 threads.
// ---------------------------------------------------------------------------
__global__ void __launch_bounds__(256)
tfm_build_matrices(const float* __restrict__ E_txt,   // (B, text_dim)
                   const float* __restrict__ Wf,      // (C*T, text_dim)
                   const float* __restrict__ bf,      // (C*T)
                   float* __restrict__ Mout,          // (B*C, 16, 16) row-major
                   int Cdim, int text_dim)
{
    __shared__ float partial[256];
    __shared__ float wgate[TDIM];
    __shared__ float Dm[TDIM][TDIM];

    const int tid = threadIdx.x;
    const int bc  = blockIdx.x;
    const int b   = bc / Cdim;
    const int c   = bc % Cdim;
    const int g   = tid >> 4;   // 0..15 : which t / which row of D / row of M
    const int l   = tid & 15;   // 0..15 : lane within group

    // --- gate GEMM row: w_row = c*16 + g ; dot(E_txt[b], Wf[row]) ---
    {
        const int row = c * TDIM + g;
        const float* __restrict__ wrow = Wf    + (size_t)row * text_dim;
        const float* __restrict__ erow = E_txt + (size_t)b   * text_dim;
        float s = 0.f;
        for (int j = l; j < text_dim; j += 16)
            s = fmaf(wrow[j], erow[j], s);
        partial[tid] = s;
    }

    // --- orthonormal DCT-II matrix in LDS: Dm[k][i] ---
    {
        const int k = g, i = l;
        float v = cosf(3.14159265358979323846f * (i + 0.5f) * (float)k / (float)TDIM)
                  * 0.35355339059327373f;              // sqrt(2/16)
        if (k == 0) v *= 0.70710678118654752f;         // 1/sqrt(2)
        Dm[k][i] = v;
    }
    __syncthreads();

    if (tid < TDIM) {
        float acc = 0.f;
        #pragma unroll
        for (int j = 0; j < 16; ++j) acc += partial[tid * 16 + j];
        acc += bf[c * TDIM + tid];
        wgate[tid] = 1.0f / (1.0f + expf(-acc));       // sigmoid
    }
    __syncthreads();

    // --- M[i][j] = sum_k D[k][i] * w[k] * D[k][j] ---
    {
        const int i = g, j = l;
        float acc = 0.f;
        #pragma unroll
        for (int k = 0; k < TDIM; ++k)
            acc = fmaf(Dm[k][i] * wgate[k], Dm[k][j], acc);
        Mout[(size_t)bc * 256 + tid] = acc;
    }
}

// ---------------------------------------------------------------------------
// Kernel 2: out[bc, :, pix] = M[bc] (16x16) @ r[bc, :, pix]  via fp32 WMMA.
// 8 waves per block; each wave owns one (bc, 64-pixel group) = 4 tiles of 16.
// A operand: M columns in 4 K-chunks (16x4 f32 -> v2f per lane).
// B operand: 4 rows of r per chunk (4x16 f32 -> v2f per lane).
// ---------------------------------------------------------------------------
__global__ void __launch_bounds__(256)
tfm_modulate_wmma(const float* __restrict__ r,
                  const float* __restrict__ M,     // (B*C, 16, 16)
                  float* __restrict__ out,
                  int HW, int tgPerBC)             // tgPerBC = HW / 64
{
    const int lane = threadIdx.x & 31;
    const int wave = threadIdx.x >> 5;
    const int wg   = blockIdx.x * 8 + wave;        // global wave id
    const int bc   = wg / tgPerBC;
    const int tg   = wg % tgPerBC;                 // 64-pixel group
    const int half = lane >> 4;                    // 0: lanes 0-15, 1: lanes 16-31
    const int lid  = lane & 15;

    // Load A (M) once, laid out for V_WMMA_F32_16X16X4_F32:
    //   VGPR0: K=2*half, VGPR1: K=2*half+1, rows striped on lanes 0-15.
    const float* __restrict__ Mb = M + (size_t)bc * 256;
    v2f a[4];
    #pragma unroll
    for (int ch = 0; ch < 4; ++ch) {
        const int col = 4 * ch + 2 * half;
        a[ch].x = Mb[lid * 16 + col];
        a[ch].y = Mb[lid * 16 + col + 1];
    }

    const float* __restrict__ rb = r   + (size_t)bc * TDIM * HW;
    float*       __restrict__ ob = out + (size_t)bc * TDIM * HW;
    const int p0base = tg * 64;

    #pragma unroll
    for (int sub = 0; sub < 4; ++sub) {
        const int p0 = p0base + sub * 16;
        v8f acc = {0.f, 0.f, 0.f, 0.f, 0.f, 0.f, 0.f, 0.f};

        #pragma unroll
        for (int ch = 0; ch < 4; ++ch) {
            // B operand: rows (K) of the input tile; mirrors the C/D half-split.
            const int rrow = 4 * ch + 2 * half;
            v2f bb;
            bb.x = rb[(size_t)(rrow    ) * HW + p0 + lid];
            bb.y = rb[(size_t)(rrow + 1) * HW + p0 + lid];
            // D = A(16x4) x B(4x16) + C, fp32 in/out
            acc = __builtin_amdgcn_wmma_f32_16x16x4_f32(
                      /*neg_a=*/false, a[ch],
                      /*neg_b=*/false, bb,
                      /*c_mod=*/(short)0, acc,
                      /*reuse_a=*/false, /*reuse_b=*/false);
        }

        // C/D layout: VGPR v -> row (v + 8*half), col = lid
        #pragma unroll
        for (int v = 0; v < 8; ++v) {
            ob[(size_t)(v + 8 * half) * HW + p0 + lid] = acc[v];
        }
    }
}

// ---------------------------------------------------------------------------
// Launch
// ---------------------------------------------------------------------------
extern "C" void kernel_launch(void* const* d_in, const int* in_sizes, int n_in,
                              void* d_out, int out_size, void* d_ws, size_t ws_size,
                              hipStream_t stream) {
    const float* r     = (const float*)d_in[0];   // (B,C,T,H,W)
    const float* E_txt = (const float*)d_in[1];   // (B, text_dim)
    const float* Wf    = (const float*)d_in[2];   // (C*T, text_dim)
    const float* bf    = (const float*)d_in[3];   // (C*T)
    float* out = (float*)d_out;

    // Derive dims (T fixed at 16 by the WMMA tiling).
    const int CT       = in_sizes[3];             // C*T = 1024
    const int text_dim = in_sizes[2] / CT;        // 768
    const int B        = in_sizes[1] / text_dim;  // 8
    const int C        = CT / TDIM;               // 64
    const int HW       = in_sizes[0] / (B * CT);  // 4096

    float* Mws = (float*)d_ws;                    // (B*C, 16, 16) = 512 KB

    // Pass 1: build 512 fused 16x16 operators.
    tfm_build_matrices<<<B * C, 256, 0, stream>>>(E_txt, Wf, bf, Mws, C, text_dim);

    // Pass 2: one read + one write of r, WMMA fp32 apply.
    const int tgPerBC = HW / 64;                  // 64
    const int nWaves  = B * C * tgPerBC;          // 32768
    const int nBlocks = nWaves / 8;               // 4096 blocks of 8 waves
    tfm_modulate_wmma<<<nBlocks, 256, 0, stream>>>(r, Mws, out, HW, tgPerBC);
}